// CrossAttentionBlock_28492813041792
// MI455X (gfx1250) — compile-verified
//
#include <hip/hip_runtime.h>
#include <hip/hip_bf16.h>
#include <math.h>

// ---------------- WMMA helpers (CDNA5 / gfx1250, wave32) ----------------
typedef __attribute__((ext_vector_type(2))) float v2f;
typedef __attribute__((ext_vector_type(8))) float v8f;

static __device__ __forceinline__ v8f wmma_f32_16x16x4(v2f a, v2f b, v8f c) {
  // D = A(16x4) * B(4x16) + C(16x16), fp32 matrix pipe
  return __builtin_amdgcn_wmma_f32_16x16x4_f32(false, a, false, b, (short)0, c,
                                               false, false);
}

// Problem constants
#define NB 8
#define CJC 256
#define CGC 256
#define TT 128
#define VV 25
#define GG 5
#define HH 8
#define DD 64
#define HD 512
#define TQ (TT * VV)   // 3200 query tokens per batch
#define SK (TT * GG)   // 640 key tokens per batch
#define BN_EPS 1e-5f

// ---------------------------------------------------------------------------
// Generic per-batch GEMM with bias:  C[b] = A(MxK) * B[b](KxN) + bias(M)
// K is a compile-time constant -> static trip count, no loop versioning.
// One wave (32 threads) computes a 32x32 output tile (2x2 register blocking
// of 16x16 WMMA tiles): 2 loads per WMMA.
// Fragment layouts (ISA 7.12.2):
//   A(m,k): lane = m + 16*(k/2), reg = k&1
//   B(k,n): lane = n + 16*(k/2), reg = k&1
//   D(m,n): reg  = m&7,  lane = n + 16*(m/8)
// ---------------------------------------------------------------------------
template <int K>
__global__ void gemm_wmma_bias(const float* __restrict__ A,
                               const float* __restrict__ B,
                               float* __restrict__ C,
                               const float* __restrict__ bias,
                               int M, int Nn) {
  const int tn = blockIdx.x;           // 32-wide output-column tile
  const int tm = blockIdx.y;           // 32-tall output-row tile
  const int b  = blockIdx.z;           // batch
  const int lane  = threadIdx.x;       // 0..31 (wave32)
  const int l15   = lane & 15;
  const int khalf = lane >> 4;         // 0 or 1

  const float* __restrict__ Bp = B + (size_t)b * (size_t)K * (size_t)Nn;
  float* __restrict__ Cp       = C + (size_t)b * (size_t)M * (size_t)Nn;

  const int m0 = tm * 32;
  const int n0 = tn * 32;

  const float* __restrict__ Arow0 = A + (size_t)(m0 + l15) * K;
  const float* __restrict__ Arow1 = A + (size_t)(m0 + 16 + l15) * K;

  v8f acc00 = {}, acc01 = {}, acc10 = {}, acc11 = {};

#pragma unroll 4
  for (int k0 = 0; k0 < K; k0 += 4) {
    const int ka = k0 + 2 * khalf;
    v2f a0, a1, b0, b1;
    a0.x = Arow0[ka];
    a0.y = Arow0[ka + 1];
    a1.x = Arow1[ka];
    a1.y = Arow1[ka + 1];
    const float* __restrict__ Br0 = Bp + (size_t)ka * Nn + n0;
    const float* __restrict__ Br1 = Bp + (size_t)(ka + 1) * Nn + n0;
    b0.x = Br0[l15];
    b0.y = Br1[l15];
    b1.x = Br0[16 + l15];
    b1.y = Br1[16 + l15];
    acc00 = wmma_f32_16x16x4(a0, b0, acc00);
    acc01 = wmma_f32_16x16x4(a0, b1, acc01);
    acc10 = wmma_f32_16x16x4(a1, b0, acc10);
    acc11 = wmma_f32_16x16x4(a1, b1, acc11);
  }

  const int mb0 = m0 + (khalf ? 8 : 0);
  const int mb1 = m0 + 16 + (khalf ? 8 : 0);
  const int nc0 = n0 + l15;
  const int nc1 = n0 + 16 + l15;
#pragma unroll
  for (int r = 0; r < 8; ++r) {
    const float bia0 = bias[mb0 + r];
    const float bia1 = bias[mb1 + r];
    Cp[(size_t)(mb0 + r) * Nn + nc0] = acc00[r] + bia0;
    Cp[(size_t)(mb0 + r) * Nn + nc1] = acc01[r] + bia0;
    Cp[(size_t)(mb1 + r) * Nn + nc0] = acc10[r] + bia1;
    Cp[(size_t)(mb1 + r) * Nn + nc1] = acc11[r] + bia1;
  }
}

// ---------------------------------------------------------------------------
// Attention: per (n, h, 16-query tile).
//   q  : [N][HD][TQ]   (head h rows h*64 .. h*64+63)
//   kv : [N][2*HD][SK] (K rows h*64.., V rows 512 + h*64..)
//   out: [N][HD][TQ]
// Pass 1: S = (Q^T K) * scale  -> LDS [16][SK]   (2-wide over key tiles)
// Softmax rows in LDS.
// Pass 2: O = V * P^T  (4-wide over d tiles; one LDS P-frag feeds 4 WMMAs)
// One wave per block; LDS 16*648*4 = 41.5 KB.
// ---------------------------------------------------------------------------
#define SLD 648   // padded LDS row stride (floats)

__global__ void attention_wmma(const float* __restrict__ q,
                               const float* __restrict__ kv,
                               float* __restrict__ out) {
  __shared__ float sc[16 * SLD];

  const int qt = blockIdx.x;   // 0..199  (16-query tile)
  const int h  = blockIdx.y;   // 0..7
  const int n  = blockIdx.z;   // 0..7
  const int lane  = threadIdx.x;
  const int l15   = lane & 15;
  const int khalf = lane >> 4;

  const float* __restrict__ Q  = q  + ((size_t)n * HD + (size_t)h * DD) * TQ;
  const float* __restrict__ Kp = kv + ((size_t)n * 2 * HD + (size_t)h * DD) * SK;
  const float* __restrict__ Vp = kv + ((size_t)n * 2 * HD + HD + (size_t)h * DD) * SK;
  float* __restrict__ O        = out + ((size_t)n * HD + (size_t)h * DD) * TQ;

  const int q0 = qt * 16;
  const float scale = 0.125f;  // 64^-0.5
  const int mrow = khalf ? 8 : 0;

  // ---- Pass 1: scores, 2 key-tiles per sweep (reuse Q fragment) ----
#pragma unroll 1
  for (int s0 = 0; s0 < SK; s0 += 32) {
    v8f acc0 = {}, acc1 = {};
#pragma unroll
    for (int k0 = 0; k0 < DD; k0 += 4) {
      const int ka = k0 + 2 * khalf;
      v2f afrag, b0, b1;
      // A(m=query, k=d) = Q[d][q0+m]
      afrag.x = Q[(size_t)ka * TQ + q0 + l15];
      afrag.y = Q[(size_t)(ka + 1) * TQ + q0 + l15];
      // B(k=d, n=key) = K[d][s0+n]
      const float* __restrict__ Kr0 = Kp + (size_t)ka * SK + s0;
      const float* __restrict__ Kr1 = Kp + (size_t)(ka + 1) * SK + s0;
      b0.x = Kr0[l15];
      b0.y = Kr1[l15];
      b1.x = Kr0[16 + l15];
      b1.y = Kr1[16 + l15];
      acc0 = wmma_f32_16x16x4(afrag, b0, acc0);
      acc1 = wmma_f32_16x16x4(afrag, b1, acc1);
    }
#pragma unroll
    for (int r = 0; r < 8; ++r) {
      sc[(mrow + r) * SLD + s0 + l15]      = acc0[r] * scale;
      sc[(mrow + r) * SLD + s0 + 16 + l15] = acc1[r] * scale;
    }
  }
  __syncthreads();

  // ---- Softmax over keys, one lane per query row (lanes 0..15) ----
  if (lane < 16) {
    float* row = &sc[lane * SLD];
    float m = -3.402823466e38f;
    for (int s = 0; s < SK; ++s) m = fmaxf(m, row[s]);
    float sum = 0.0f;
    for (int s = 0; s < SK; ++s) {
      float e = __expf(row[s] - m);
      row[s] = e;
      sum += e;
    }
    const float inv = 1.0f / sum;
    for (int s = 0; s < SK; ++s) row[s] *= inv;
  }
  __syncthreads();

  // ---- Pass 2: O(d, q) = sum_s V(d,s) * P(q,s), all 4 d-tiles at once ----
  v8f oacc0 = {}, oacc1 = {}, oacc2 = {}, oacc3 = {};
  const float* __restrict__ V0 = Vp + (size_t)(l15) * SK;
  const float* __restrict__ V1 = Vp + (size_t)(16 + l15) * SK;
  const float* __restrict__ V2 = Vp + (size_t)(32 + l15) * SK;
  const float* __restrict__ V3 = Vp + (size_t)(48 + l15) * SK;
#pragma unroll 2
  for (int s0 = 0; s0 < SK; s0 += 4) {
    const int ka = s0 + 2 * khalf;
    v2f bfrag;
    // B(k=s, n=q_local) = P(q, s) = sc[q*SLD + s]  (one read, four uses)
    bfrag.x = sc[l15 * SLD + ka];
    bfrag.y = sc[l15 * SLD + ka + 1];
    v2f a0, a1, a2, a3;
    a0.x = V0[ka]; a0.y = V0[ka + 1];
    a1.x = V1[ka]; a1.y = V1[ka + 1];
    a2.x = V2[ka]; a2.y = V2[ka + 1];
    a3.x = V3[ka]; a3.y = V3[ka + 1];
    oacc0 = wmma_f32_16x16x4(a0, bfrag, oacc0);
    oacc1 = wmma_f32_16x16x4(a1, bfrag, oacc1);
    oacc2 = wmma_f32_16x16x4(a2, bfrag, oacc2);
    oacc3 = wmma_f32_16x16x4(a3, bfrag, oacc3);
  }
#pragma unroll
  for (int r = 0; r < 8; ++r) {
    O[(size_t)(mrow + r) * TQ + q0 + l15]      = oacc0[r];
    O[(size_t)(16 + mrow + r) * TQ + q0 + l15] = oacc1[r];
    O[(size_t)(32 + mrow + r) * TQ + q0 + l15] = oacc2[r];
    O[(size_t)(48 + mrow + r) * TQ + q0 + l15] = oacc3[r];
  }
}

// ---------------------------------------------------------------------------
// BatchNorm statistics: one block per channel o (256 channels).
// x: [N][CJC][TQ]; stats[o] = mean, stats[256+o] = var (biased).
// ---------------------------------------------------------------------------
__global__ void bn_stats(const float* __restrict__ x, float* __restrict__ stats) {
  __shared__ float ssum[256];
  __shared__ float ssq[256];
  const int o   = blockIdx.x;
  const int tid = threadIdx.x;
  float sum = 0.0f, sq = 0.0f;
  const int total = NB * TQ;  // 25600
  for (int i = tid; i < total; i += blockDim.x) {
    const int nb = i / TQ;
    const int p  = i - nb * TQ;
    const float v = x[((size_t)nb * CJC + o) * TQ + p];
    sum += v;
    sq  += v * v;
  }
  ssum[tid] = sum;
  ssq[tid]  = sq;
  __syncthreads();
  for (int s = 128; s > 0; s >>= 1) {
    if (tid < s) {
      ssum[tid] += ssum[tid + s];
      ssq[tid]  += ssq[tid + s];
    }
    __syncthreads();
  }
  if (tid == 0) {
    const float inv = 1.0f / (float)total;
    const float mean = ssum[0] * inv;
    const float var  = ssq[0] * inv - mean * mean;
    stats[o]       = mean;
    stats[256 + o] = var;
  }
}

// ---------------------------------------------------------------------------
// BatchNorm apply: out = (x - mean) * rsqrt(var+eps) * gamma + beta
// ---------------------------------------------------------------------------
__global__ void bn_apply(const float* __restrict__ x,
                         const float* __restrict__ stats,
                         const float* __restrict__ gamma,
                         const float* __restrict__ beta,
                         float* __restrict__ out) {
  const size_t idx = (size_t)blockIdx.x * blockDim.x + threadIdx.x;
  const size_t total = (size_t)NB * CJC * TQ;
  if (idx >= total) return;
  const int o = (int)((idx / TQ) % CJC);
  const float mean = stats[o];
  const float var  = stats[256 + o];
  out[idx] = (x[idx] - mean) * rsqrtf(var + BN_EPS) * gamma[o] + beta[o];
}

// ---------------------------------------------------------------------------
extern "C" void kernel_launch(void* const* d_in, const int* in_sizes, int n_in,
                              void* d_out, int out_size, void* d_ws, size_t ws_size,
                              hipStream_t stream) {
  const float* joint = (const float*)d_in[0];  // [8][256][3200]
  const float* group = (const float*)d_in[1];  // [8][256][640]
  const float* wq    = (const float*)d_in[2];  // [512][256]
  const float* bq    = (const float*)d_in[3];  // [512]
  const float* wkv   = (const float*)d_in[4];  // [1024][256]
  const float* bkv   = (const float*)d_in[5];  // [1024]
  const float* wo    = (const float*)d_in[6];  // [256][512]
  const float* bo    = (const float*)d_in[7];  // [256]
  const float* gamma = (const float*)d_in[8];  // [256]
  const float* beta  = (const float*)d_in[9];  // [256]
  float* outp = (float*)d_out;

  float* ws = (float*)d_ws;
  const size_t off_q  = 0;                                       // [8][512][3200]
  const size_t off_kv = off_q  + (size_t)NB * HD * TQ;           // [8][1024][640]
  const size_t off_ao = off_kv + (size_t)NB * 2 * HD * SK;       // [8][512][3200]
  const size_t off_o2 = off_ao + (size_t)NB * HD * TQ;           // [8][256][3200]
  const size_t off_st = off_o2 + (size_t)NB * CJC * TQ;          // [512]

  float* q_ws  = ws + off_q;
  float* kv_ws = ws + off_kv;
  float* ao_ws = ws + off_ao;
  float* o2_ws = ws + off_o2;
  float* st_ws = ws + off_st;

  // 1) Q = wq * joint + bq          (M=512, N=3200, K=256 per batch)
  gemm_wmma_bias<CJC><<<dim3(TQ / 32, HD / 32, NB), 32, 0, stream>>>(
      wq, joint, q_ws, bq, HD, TQ);

  // 2) KV = wkv * group + bkv       (M=1024, N=640, K=256 per batch)
  gemm_wmma_bias<CGC><<<dim3(SK / 32, (2 * HD) / 32, NB), 32, 0, stream>>>(
      wkv, group, kv_ws, bkv, 2 * HD, SK);

  // 3) Attention per (n, h, 16-query tile)
  attention_wmma<<<dim3(TQ / 16, HH, NB), 32, 0, stream>>>(q_ws, kv_ws, ao_ws);

  // 4) Out = wo * attn_out + bo     (M=256, N=3200, K=512 per batch)
  gemm_wmma_bias<HD><<<dim3(TQ / 32, CJC / 32, NB), 32, 0, stream>>>(
      wo, ao_ws, o2_ws, bo, CJC, TQ);

  // 5) BatchNorm statistics (256 channels)
  bn_stats<<<dim3(CJC), 256, 0, stream>>>(o2_ws, st_ws);

  // 6) BatchNorm apply -> d_out
  const size_t total = (size_t)NB * CJC * TQ;
  bn_apply<<<dim3((unsigned)((total + 255) / 256)), 256, 0, stream>>>(
      o2_ws, st_ws, gamma, beta, outp);

  (void)in_sizes; (void)n_in; (void)out_size; (void)ws_size;
}